// BondGATMessagePassing_88914412961897
// MI455X (gfx1250) — compile-verified
//
#include <hip/hip_runtime.h>
#include <hip/hip_bf16.h>
#include <math.h>

#define NN     50000      // nodes
#define EE     800000     // edges
#define NF     128        // node features
#define BF     16         // bond features
#define HH     4          // heads
#define CC     32         // per-head channels
#define HC     128        // HH*CC
#define DEPTH  5
#define NEG_SLOPE 0.2f

typedef __attribute__((ext_vector_type(2))) float v2f;
typedef __attribute__((ext_vector_type(8))) float v8f;

// ---------- helpers ----------
__device__ __forceinline__ float sanitize_f(float v) {
    if (isnan(v)) return 0.0f;
    if (isinf(v)) return v > 0.0f ? 1000.0f : -1000.0f;
    return v;
}
// order-preserving float -> uint key (for atomicMax-based segment max)
__device__ __forceinline__ unsigned fkey(float f) {
    unsigned u = __float_as_uint(f);
    return (u & 0x80000000u) ? ~u : (u | 0x80000000u);
}
__device__ __forceinline__ float funkey(unsigned k) {
    unsigned u = (k & 0x80000000u) ? (k & 0x7FFFFFFFu) : ~k;
    return __uint_as_float(u);
}

// ---------- init: h = sanitize(x) ----------
__global__ void init_h_kernel(const float* __restrict__ x, float* __restrict__ h) {
    long long i = (long long)blockIdx.x * blockDim.x + threadIdx.x;
    if (i < (long long)NN * NF) h[i] = sanitize_f(x[i]);
}

// ---------- zero deg + self-loop attr accumulators ----------
__global__ void zero_degsl_kernel(float* __restrict__ deg, float* __restrict__ sl) {
    long long i = (long long)blockIdx.x * blockDim.x + threadIdx.x;
    if (i < (long long)NN) deg[i] = 0.0f;
    if (i < (long long)NN * BF) sl[i] = 0.0f;
}

// ---------- accumulate deg & sum(edge_attr) by dst ----------
__global__ void degsl_acc_kernel(const int* __restrict__ dst,
                                 const float* __restrict__ ea,
                                 float* __restrict__ deg, float* __restrict__ sl) {
    long long g = (long long)blockIdx.x * blockDim.x + threadIdx.x;
    if (g >= (long long)EE * BF) return;
    int e = (int)(g >> 4);
    int f = (int)(g & (BF - 1));
    int d = dst[e];
    atomicAdd(&sl[(size_t)d * BF + f], ea[(size_t)e * BF + f]);
    if (f == 0) atomicAdd(&deg[d], 1.0f);
}

__global__ void sl_norm_kernel(const float* __restrict__ deg, float* __restrict__ sl) {
    long long g = (long long)blockIdx.x * blockDim.x + threadIdx.x;
    if (g >= (long long)NN * BF) return;
    int n = (int)(g >> 4);
    sl[g] /= fmaxf(deg[n], 1.0f);
}

// ---------- fold attention vectors: vs,vd [128][H], ve [16][H] ----------
__global__ void fold_kernel(const float* __restrict__ Wl,  const float* __restrict__ asl,
                            const float* __restrict__ adl, const float* __restrict__ Wel,
                            const float* __restrict__ ael,
                            float* __restrict__ vs, float* __restrict__ vd,
                            float* __restrict__ ve) {
    int k = threadIdx.x;          // 0..127
    if (k < HC) {
        #pragma unroll
        for (int h = 0; h < HH; ++h) {
            float s0 = 0.f, s1 = 0.f;
            #pragma unroll 8
            for (int c = 0; c < CC; ++c) {
                float w = Wl[k * HC + h * CC + c];
                s0 += w * asl[h * CC + c];
                s1 += w * adl[h * CC + c];
            }
            vs[k * HH + h] = s0;
            vd[k * HH + h] = s1;
        }
    }
    if (k < BF) {
        #pragma unroll
        for (int h = 0; h < HH; ++h) {
            float s = 0.f;
            #pragma unroll 8
            for (int c = 0; c < CC; ++c)
                s += Wel[k * HC + h * CC + c] * ael[h * CC + c];
            ve[k * HH + h] = s;
        }
    }
}

// ---------- WMMA GEMM: xp = h @ W  ([N,128] x [128,128]) ----------
// 256 threads = 8 waves; W staged in 64KB LDS; each wave -> one 16-row strip,
// 8 accumulators covering all 128 output columns; K loop of 32 x V_WMMA_F32_16X16X4_F32.
__global__ void __launch_bounds__(256)
gemm_node_wmma_kernel(const float* __restrict__ hsrc, const float* __restrict__ Wl,
                      float* __restrict__ xp) {
    __shared__ float lW[HC * HC];               // 64 KB
    const float4* Wv = (const float4*)Wl;
    float4* lWv = (float4*)lW;
    for (int i = threadIdx.x; i < HC * HC / 4; i += blockDim.x) lWv[i] = Wv[i];
    __syncthreads();

    const int wave = threadIdx.x >> 5;
    const int lane = threadIdx.x & 31;
    const int strip = blockIdx.x * 8 + wave;
    if (strip >= NN / 16) return;               // 50000 % 16 == 0

    const int row0 = strip * 16;
    const int lm = lane & 15;
    const int hi = lane >> 4;                   // 0: K+{0,1}, 1: K+{2,3}

    v8f acc[8] = {};                            // 8 tiles x 16 cols = 128 cols
    const float* __restrict__ arow = hsrc + (size_t)(row0 + lm) * HC;

    for (int k0 = 0; k0 < HC; k0 += 4) {
        const int kk = k0 + hi * 2;
        __builtin_prefetch(arow + k0 + 16, 0, 1);      // global_prefetch_b8
        v2f a;
        a.x = arow[kk];
        a.y = arow[kk + 1];
        #pragma unroll
        for (int t = 0; t < 8; ++t) {
            const int ncol = t * 16 + lm;
            v2f b;
            b.x = lW[kk * HC + ncol];
            b.y = lW[(kk + 1) * HC + ncol];
            acc[t] = __builtin_amdgcn_wmma_f32_16x16x4_f32(
                false, a, false, b, (short)0, acc[t], false, false);
        }
    }
    // C/D layout: VGPR r -> rows row0+r (lanes 0-15) and row0+8+r (lanes 16-31)
    #pragma unroll
    for (int r = 0; r < 8; ++r) {
        const int m = row0 + r + hi * 8;
        float* __restrict__ orow = xp + (size_t)m * HC + lm;
        #pragma unroll
        for (int t = 0; t < 8; ++t) orow[t * 16] = acc[t][r];
    }
}

// ---------- per-node attention scalars + softmax state init ----------
__global__ void node_attn_kernel(const float* __restrict__ h,
                                 const float* __restrict__ vs, const float* __restrict__ vd,
                                 float* __restrict__ a_s, float* __restrict__ a_d,
                                 unsigned* __restrict__ mkey, float* __restrict__ den) {
    __shared__ float svs[HC * HH];
    __shared__ float svd[HC * HH];
    for (int i = threadIdx.x; i < HC * HH; i += blockDim.x) { svs[i] = vs[i]; svd[i] = vd[i]; }
    __syncthreads();
    int n = blockIdx.x * blockDim.x + threadIdx.x;
    if (n >= NN) return;
    float as[HH] = {0, 0, 0, 0}, ad[HH] = {0, 0, 0, 0};
    const float* hr = h + (size_t)n * HC;
    for (int k = 0; k < HC; ++k) {
        float hv = hr[k];
        #pragma unroll
        for (int t = 0; t < HH; ++t) {
            as[t] += hv * svs[k * HH + t];
            ad[t] += hv * svd[k * HH + t];
        }
    }
    #pragma unroll
    for (int t = 0; t < HH; ++t) {
        a_s[n * HH + t] = as[t];
        a_d[n * HH + t] = ad[t];
        mkey[n * HH + t] = 0u;      // -> -inf under fkey ordering
        den[n * HH + t] = 0.0f;
    }
}

// ---------- edge alpha + segment max ----------
__global__ void edge_alpha_kernel(const int* __restrict__ src, const int* __restrict__ dst,
                                  const float* __restrict__ ea, const float* __restrict__ sl,
                                  const float* __restrict__ ve,
                                  const float* __restrict__ a_s, const float* __restrict__ a_d,
                                  float* __restrict__ alpha, unsigned* __restrict__ mkey) {
    __shared__ float sve[BF * HH];
    for (int i = threadIdx.x; i < BF * HH; i += blockDim.x) sve[i] = ve[i];
    __syncthreads();
    long long e = (long long)blockIdx.x * blockDim.x + threadIdx.x;
    if (e >= (long long)(EE + NN)) return;
    int s, d; const float* ap;
    if (e < EE) { s = src[e]; d = dst[e]; ap = ea + (size_t)e * BF; }
    else        { s = d = (int)(e - EE); ap = sl + (size_t)(e - EE) * BF; }
    float aev[HH] = {0, 0, 0, 0};
    #pragma unroll 4
    for (int f = 0; f < BF; ++f) {
        float av = ap[f];
        #pragma unroll
        for (int t = 0; t < HH; ++t) aev[t] += av * sve[f * HH + t];
    }
    #pragma unroll
    for (int t = 0; t < HH; ++t) {
        float al = a_s[s * HH + t] + a_d[d * HH + t] + aev[t];
        al = al > 0.0f ? al : NEG_SLOPE * al;
        alpha[e * HH + t] = al;
        atomicMax(&mkey[d * HH + t], fkey(al));
    }
}

// ---------- exp(alpha - max) + segment sum ----------
__global__ void edge_exp_kernel(const int* __restrict__ dst,
                                const unsigned* __restrict__ mkey,
                                float* __restrict__ alpha, float* __restrict__ den) {
    long long g = (long long)blockIdx.x * blockDim.x + threadIdx.x;
    if (g >= (long long)(EE + NN) * HH) return;
    long long e = g >> 2;
    int t = (int)(g & 3);
    int d = (e < EE) ? dst[e] : (int)(e - EE);
    float ex = expf(alpha[g] - funkey(mkey[d * HH + t]));
    alpha[g] = ex;
    atomicAdd(&den[d * HH + t], ex);
}

// ---------- hn = bias (broadcast) ----------
__global__ void init_hn_kernel(const float* __restrict__ bl, float* __restrict__ hn) {
    long long i = (long long)blockIdx.x * blockDim.x + threadIdx.x;
    if (i < (long long)NN * HC) hn[i] = bl[i & (HC - 1)];
}

// ---------- aggregation: hn[d] += coef * xp[s] (128 threads per edge) ----------
__global__ void aggregate_kernel(const int* __restrict__ src, const int* __restrict__ dst,
                                 const float* __restrict__ ex, const float* __restrict__ den,
                                 const float* __restrict__ xp, float* __restrict__ hn) {
    long long g = (long long)blockIdx.x * blockDim.x + threadIdx.x;
    long long e = g >> 7;
    if (e >= (long long)(EE + NN)) return;
    int j = (int)(g & 127);
    int s, d;
    if (e < EE) { s = src[e]; d = dst[e]; }
    else        { s = d = (int)(e - EE); }
    int t = j >> 5;                               // head = j / C
    float coef = ex[e * HH + t] / (den[d * HH + t] + 1e-16f);
    atomicAdd(&hn[(size_t)d * HC + j], coef * xp[(size_t)s * HC + j]);
}

// ---------- relu + sanitize in place ----------
__global__ void finalize_kernel(float* __restrict__ hn) {
    long long i = (long long)blockIdx.x * blockDim.x + threadIdx.x;
    if (i >= (long long)NN * HC) return;
    float v = hn[i];
    v = fmaxf(v, 0.0f);
    hn[i] = sanitize_f(v);
}

// ---------- output: relu([x,h] @ Wout + bout), sanitized ----------
__global__ void out_kernel(const float* __restrict__ x, const float* __restrict__ h,
                           const float* __restrict__ Wout, const float* __restrict__ bout,
                           float* __restrict__ out) {
    __shared__ float sW[(NF + HC) * CC];          // 32 KB
    __shared__ float sb[CC];
    for (int i = threadIdx.x; i < (NF + HC) * CC; i += blockDim.x) sW[i] = Wout[i];
    if (threadIdx.x < CC) sb[threadIdx.x] = bout[threadIdx.x];
    __syncthreads();
    long long g = (long long)blockIdx.x * blockDim.x + threadIdx.x;
    if (g >= (long long)NN * CC) return;
    int n = (int)(g >> 5);
    int c = (int)(g & 31);
    float acc = sb[c];
    const float* xr = x + (size_t)n * NF;
    const float* hr = h + (size_t)n * HC;
    #pragma unroll 4
    for (int k = 0; k < NF; ++k) acc += xr[k] * sW[k * CC + c];
    #pragma unroll 4
    for (int k = 0; k < HC; ++k) acc += hr[k] * sW[(NF + k) * CC + c];
    out[g] = sanitize_f(fmaxf(acc, 0.0f));
}

static inline long long cdiv(long long a, long long b) { return (a + b - 1) / b; }

extern "C" void kernel_launch(void* const* d_in, const int* in_sizes, int n_in,
                              void* d_out, int out_size, void* d_ws, size_t ws_size,
                              hipStream_t stream) {
    (void)in_sizes; (void)n_in; (void)out_size; (void)ws_size;
    const float* x        = (const float*)d_in[0];   // [N,128]
    const int*   eidx     = (const int*)  d_in[1];   // [2,E]
    const float* ea       = (const float*)d_in[2];   // [E,16]
    const float* W        = (const float*)d_in[3];   // [5,128,128]
    const float* att_src  = (const float*)d_in[4];   // [5,4,32]
    const float* att_dst  = (const float*)d_in[5];
    const float* We       = (const float*)d_in[6];   // [5,16,128]
    const float* att_edge = (const float*)d_in[7];
    const float* b        = (const float*)d_in[8];   // [5,128]
    const float* Wout     = (const float*)d_in[9];   // [256,32]
    const float* bout     = (const float*)d_in[10];  // [32]
    float* out = (float*)d_out;

    const int* src = eidx;
    const int* dst = eidx + EE;

    // workspace carve (floats)
    float* ws = (float*)d_ws;
    size_t off = 0;
    float* hA    = ws + off; off += (size_t)NN * HC;
    float* hB    = ws + off; off += (size_t)NN * HC;
    float* xp    = ws + off; off += (size_t)NN * HC;
    float* a_s   = ws + off; off += (size_t)NN * HH;
    float* a_d   = ws + off; off += (size_t)NN * HH;
    float* alpha = ws + off; off += (size_t)(EE + NN) * HH;
    float* mraw  = ws + off; off += (size_t)NN * HH;
    float* den   = ws + off; off += (size_t)NN * HH;
    float* sl    = ws + off; off += (size_t)NN * BF;
    float* deg   = ws + off; off += (size_t)NN;
    float* vs    = ws + off; off += (size_t)HC * HH;
    float* vd    = ws + off; off += (size_t)HC * HH;
    float* ve    = ws + off; off += (size_t)BF * HH;
    unsigned* mkey = (unsigned*)mraw;

    const int B256 = 256;

    // self-loop attrs + initial h
    init_h_kernel<<<(unsigned)cdiv((long long)NN * NF, B256), B256, 0, stream>>>(x, hA);
    zero_degsl_kernel<<<(unsigned)cdiv((long long)NN * BF, B256), B256, 0, stream>>>(deg, sl);
    degsl_acc_kernel<<<(unsigned)cdiv((long long)EE * BF, B256), B256, 0, stream>>>(dst, ea, deg, sl);
    sl_norm_kernel<<<(unsigned)cdiv((long long)NN * BF, B256), B256, 0, stream>>>(deg, sl);

    float* hcur = hA;
    float* hnxt = hB;
    for (int l = 0; l < DEPTH; ++l) {
        const float* Wl  = W        + (size_t)l * HC * NF;
        const float* asl = att_src  + (size_t)l * HH * CC;
        const float* adl = att_dst  + (size_t)l * HH * CC;
        const float* Wel = We       + (size_t)l * BF * HC;
        const float* ael = att_edge + (size_t)l * HH * CC;
        const float* bl  = b        + (size_t)l * HC;

        fold_kernel<<<1, 128, 0, stream>>>(Wl, asl, adl, Wel, ael, vs, vd, ve);
        gemm_node_wmma_kernel<<<(unsigned)cdiv(NN / 16, 8), 256, 0, stream>>>(hcur, Wl, xp);
        node_attn_kernel<<<(unsigned)cdiv(NN, B256), B256, 0, stream>>>(
            hcur, vs, vd, a_s, a_d, mkey, den);
        edge_alpha_kernel<<<(unsigned)cdiv(EE + NN, B256), B256, 0, stream>>>(
            src, dst, ea, sl, ve, a_s, a_d, alpha, mkey);
        edge_exp_kernel<<<(unsigned)cdiv((long long)(EE + NN) * HH, B256), B256, 0, stream>>>(
            dst, mkey, alpha, den);
        init_hn_kernel<<<(unsigned)cdiv((long long)NN * HC, B256), B256, 0, stream>>>(bl, hnxt);
        aggregate_kernel<<<(unsigned)cdiv((long long)(EE + NN) * HC, B256), B256, 0, stream>>>(
            src, dst, alpha, den, xp, hnxt);
        finalize_kernel<<<(unsigned)cdiv((long long)NN * HC, B256), B256, 0, stream>>>(hnxt);

        float* tmp = hcur; hcur = hnxt; hnxt = tmp;
    }

    out_kernel<<<(unsigned)cdiv((long long)NN * CC, B256), B256, 0, stream>>>(
        x, hcur, Wout, bout, out);
}